// CPIC_31207232372866
// MI455X (gfx1250) — compile-verified
//
#include <hip/hip_runtime.h>
#include <hip/hip_bf16.h>
#include <math.h>

typedef float v2f __attribute__((ext_vector_type(2)));
typedef float v8f __attribute__((ext_vector_type(8)));

#define B_N 1024
#define D_N 256
#define THRESH 1e-6f
#define NSPLIT 4           // column splits per 16-row strip
#define TILES_PER_SPLIT 16 // 64 j-tiles / NSPLIT
#define NWAVES 8
#define TPW 2              // tiles per wave = TILES_PER_SPLIT / NWAVES

// workspace layout (float offsets)
#define OFF_INVV 0         // [B_N*D_N]        1/v'
#define OFF_AMAT 262144    // [B_N*D_N]        m/v'
#define OFF_CJ   524288    // [B_N]            sum(log v' + m^2/v')
#define OFF_PART 525312    // [B_N][NSPLIT][4] (m, s, m_nd, s_nd)
#define OFF_DIAG 541696    // [B_N]            diag scores

// logsumexp merge of (m,s) with (om,os): represents log(s) + m
__device__ __forceinline__ void lse_combine(float& m, float& s, float om, float os) {
    const float nm = fmaxf(m, om);
    s = s * __expf(m - nm) + os * __expf(om - nm);
    m = nm;
}

// ---------------------------------------------------------------- kernel 1
__global__ __launch_bounds__(256) void CPIC_prep(const float* __restrict__ xm,
                                                 const float* __restrict__ xv,
                                                 float* __restrict__ invv,
                                                 float* __restrict__ amat,
                                                 float* __restrict__ Cj) {
    const int j = blockIdx.x;
    const int d = threadIdx.x;
    const int idx = j * D_N + d;
    const float m = xm[idx];
    const float v = xv[idx];
    const float vp = (v < THRESH) ? v + THRESH : v;
    const float iv = 1.0f / vp;
    invv[idx] = iv;
    amat[idx] = m * iv;

    __shared__ float red[256];
    red[d] = __logf(vp) + m * m * iv;
    __syncthreads();
    for (int s = 128; s > 0; s >>= 1) {
        if (d < s) red[d] += red[d + s];
        __syncthreads();
    }
    if (d == 0) Cj[j] = red[0];
}

// ---------------------------------------------------------------- kernel 2
// Fused WMMA GEMM + two-pass row logsumexp (scores stay in registers).
// Block = (rowblk, split): rows [rowblk*16, +16), j-tiles [split*16, +16).
__global__ __launch_bounds__(256) void CPIC_gemm_lse(const float* __restrict__ y,
                                                     const float* __restrict__ invv,
                                                     const float* __restrict__ amat,
                                                     const float* __restrict__ Cj,
                                                     float* __restrict__ part,
                                                     float* __restrict__ diag) {
    const int rowblk = blockIdx.x >> 2;  // 0..63
    const int sp     = blockIdx.x & 3;   // 0..3
    const int wave = threadIdx.x >> 5;
    const int lane = threadIdx.x & 31;
    const int h   = lane >> 4;
    const int l16 = lane & 15;

    const float* __restrict__ yrow = y + (rowblk * 16 + l16) * D_N;  // A: M = l16
    const int koff = 2 * h;  // A & B: per-lane K = k + 2h + r

    __shared__ float pm[NWAVES][16], ps[NWAVES][16], pmn[NWAVES][16], psn[NWAVES][16];
    __shared__ float sdiag[16];

    float sc[TPW][8];    // scores, D: M = r + 8h, N = tile col l16
    float snd[TPW][8];   // diag-masked copy

#pragma unroll
    for (int t = 0; t < TPW; ++t) {
        const int tj = sp * TILES_PER_SPLIT + wave + NWAVES * t;
        const float* __restrict__ ivrow = invv + (tj * 16 + l16) * D_N;  // B: N = l16
        const float* __restrict__ arow  = amat + (tj * 16 + l16) * D_N;

        v8f acc1 = {};  // Ysq @ invV^T
        v8f acc2 = {};  // Y   @ (m/v)^T
#pragma unroll 4
        for (int k = 0; k < D_N; k += 4) {
            v2f yv  = *(const v2f*)(yrow  + k + koff);
            v2f b1  = *(const v2f*)(ivrow + k + koff);
            v2f b2  = *(const v2f*)(arow  + k + koff);
            v2f ysq = yv * yv;
            acc1 = __builtin_amdgcn_wmma_f32_16x16x4_f32(false, ysq, false, b1,
                                                         (short)0, acc1, false, false);
            acc2 = __builtin_amdgcn_wmma_f32_16x16x4_f32(false, yv, false, b2,
                                                         (short)0, acc2, false, false);
        }

        const float cj = Cj[tj * 16 + l16];
        const bool diagTile = (tj == rowblk);
#pragma unroll
        for (int r = 0; r < 8; ++r) {
            const float s = -0.5f * (cj + acc1[r] - 2.0f * acc2[r]);
            sc[t][r] = s;
            snd[t][r] = (diagTile && (l16 == r + 8 * h)) ? -INFINITY : s;
        }
        // diag capture: select chain + one divergent store (no per-r exec churn)
        if (diagTile) {
            float dv = sc[t][0];
#pragma unroll
            for (int r = 1; r < 8; ++r) dv = (l16 == r + 8 * h) ? sc[t][r] : dv;
            if ((l16 >> 3) == h) sdiag[l16] = dv;
        }
    }

    // two-pass LSE per row: (1) pure max reduce, (2) exp + pure sum reduce.
    float mx[8], sm[8], mxn[8], smn[8];
#pragma unroll
    for (int r = 0; r < 8; ++r) {
        float mf = fmaxf(sc[0][r], sc[1][r]);
        float mn = fmaxf(snd[0][r], snd[1][r]);
#pragma unroll
        for (int off = 1; off < 16; off <<= 1) {
            mf = fmaxf(mf, __shfl_xor(mf, off, 32));
            mn = fmaxf(mn, __shfl_xor(mn, off, 32));
        }
        float sf = __expf(sc[0][r] - mf)  + __expf(sc[1][r] - mf);
        float sn = __expf(snd[0][r] - mn) + __expf(snd[1][r] - mn);  // exp(-inf)=0 masks diag
#pragma unroll
        for (int off = 1; off < 16; off <<= 1) {
            sf += __shfl_xor(sf, off, 32);
            sn += __shfl_xor(sn, off, 32);
        }
        mx[r] = mf; sm[r] = sf; mxn[r] = mn; smn[r] = sn;
    }

    if (l16 == 0) {
#pragma unroll
        for (int r = 0; r < 8; ++r) {
            const int row = r + 8 * h;
            pm[wave][row]  = mx[r];  ps[wave][row]  = sm[r];
            pmn[wave][row] = mxn[r]; psn[wave][row] = smn[r];
        }
    }
    __syncthreads();

    if (threadIdx.x < 16) {
        const int row = threadIdx.x;
        float m = -INFINITY, s = 0.0f, mn = -INFINITY, sn = 0.0f;
        for (int w = 0; w < NWAVES; ++w) {
            lse_combine(m,  s,  pm[w][row],  ps[w][row]);
            lse_combine(mn, sn, pmn[w][row], psn[w][row]);
        }
        const int grow = rowblk * 16 + row;
        float* p = part + (grow * NSPLIT + sp) * 4;
        p[0] = m; p[1] = s; p[2] = mn; p[3] = sn;
        if (sp == (rowblk >> 4)) diag[grow] = sdiag[row];  // diag tile lives in this split
    }
}

// ---------------------------------------------------------------- kernel 3
__global__ __launch_bounds__(256) void CPIC_final(const float* __restrict__ part,
                                                  const float* __restrict__ diag,
                                                  float* __restrict__ out) {
    const int t = threadIdx.x;
    float lo = 0.0f, hi = 0.0f;
#pragma unroll
    for (int q = 0; q < 4; ++q) {
        const int i = t + q * 256;
        float m = -INFINITY, s = 0.0f, mn = -INFINITY, sn = 0.0f;
        for (int sp = 0; sp < NSPLIT; ++sp) {
            const float* p = part + (i * NSPLIT + sp) * 4;
            lse_combine(m,  s,  p[0], p[1]);
            lse_combine(mn, sn, p[2], p[3]);
        }
        const float d = diag[i];
        lo += d - (m + __logf(s));    // diag - logsumexp(row)
        hi += d - (mn + __logf(sn));  // diag - logsumexp(row \ diag)
    }

    __shared__ float r1[256], r2[256];
    r1[t] = lo;
    r2[t] = hi;
    __syncthreads();
    for (int st = 128; st > 0; st >>= 1) {
        if (t < st) { r1[t] += r1[t + st]; r2[t] += r2[t + st]; }
        __syncthreads();
    }
    if (t == 0) {
        out[0] = logf((float)B_N) + r1[0] * (1.0f / B_N);        // mi_lower
        out[1] = r2[0] * (1.0f / B_N) + logf((float)(B_N - 1));  // mi_upper
    }
}

extern "C" void kernel_launch(void* const* d_in, const int* in_sizes, int n_in,
                              void* d_out, int out_size, void* d_ws, size_t ws_size,
                              hipStream_t stream) {
    const float* x_mean = (const float*)d_in[0];
    const float* x_vars = (const float*)d_in[1];
    const float* y      = (const float*)d_in[2];
    float* out = (float*)d_out;
    float* ws  = (float*)d_ws;

    float* invv = ws + OFF_INVV;
    float* amat = ws + OFF_AMAT;
    float* Cj   = ws + OFF_CJ;
    float* part = ws + OFF_PART;
    float* diag = ws + OFF_DIAG;

    CPIC_prep<<<B_N, 256, 0, stream>>>(x_mean, x_vars, invv, amat, Cj);
    CPIC_gemm_lse<<<(B_N / 16) * NSPLIT, 256, 0, stream>>>(y, invv, amat, Cj, part, diag);
    CPIC_final<<<1, 256, 0, stream>>>(part, diag, out);
}